// AttentionGNNLayer_22548578304602
// MI455X (gfx1250) — compile-verified
//
#include <hip/hip_runtime.h>

typedef __attribute__((ext_vector_type(2))) float v2f;
typedef __attribute__((ext_vector_type(8))) float v8f;

#define NNODES 8192
#define DDIM   256   // D_IN == D_OUT == 256

// ---------------------------------------------------------------------------
// Kernel 0: zero the output (d_out is poisoned before timing, scatter needs 0)
// ---------------------------------------------------------------------------
__global__ __launch_bounds__(256)
void zero_kernel(float* __restrict__ out, int n) {
    int i = blockIdx.x * blockDim.x + threadIdx.x;
    if (i < n) out[i] = 0.0f;
}

// ---------------------------------------------------------------------------
// Kernel 1: fused triple GEMM via V_WMMA_F32_16X16X4_F32 (full f32 precision)
//   z=0: h_proj = h @ W          -> ws[0 .. NNODES*DDIM)
//   z=1: q      = h @ Wq + bq    -> ws[NNODES*DDIM .. 2*NNODES*DDIM)
//   z=2: k      = h @ Wk + bk    -> ws[2*NNODES*DDIM .. 3*NNODES*DDIM)
// One wave computes a 16x64 strip: A fragment reused across 4 WMMAs per K-step.
//
// ISA layouts (CDNA5 ISA):
//   A 16x4 f32 : lane m = lane&15, v2f = {A[m][kb], A[m][kb+1]}, kb = (lane>>4)*2
//   B  4x16 f32: lane n = lane&15, v2f = {B[kb][n], B[kb+1][n]}
//   C/D 16x16  : VGPR v -> M = v + (lane>>4)*8, N = lane&15
// ---------------------------------------------------------------------------
__global__ __launch_bounds__(32)
void gemm3_wmma_f32(const float* __restrict__ h,
                    const float* __restrict__ W,
                    const float* __restrict__ Wq, const float* __restrict__ bq,
                    const float* __restrict__ Wk, const float* __restrict__ bk,
                    float* __restrict__ ws) {
    const int z = blockIdx.z;
    const float* B    = (z == 0) ? W       : (z == 1) ? Wq : Wk;
    const float* bias = (z == 0) ? nullptr : (z == 1) ? bq : bk;
    float* C = ws + (size_t)z * NNODES * DDIM;

    const int lane = threadIdx.x & 31;
    const int m0   = blockIdx.y * 16;          // row-tile base (0..8191 step 16)
    const int n0   = blockIdx.x * 64;          // col-strip base (0..255 step 64)
    const int nc   = lane & 15;
    const int kb   = (lane >> 4) * 2;          // 0 for lanes 0-15, 2 for lanes 16-31
    const int mrow = m0 + nc;

    const float* Arow = h + (size_t)mrow * DDIM;

    v8f acc0 = {}, acc1 = {}, acc2 = {}, acc3 = {};

    for (int k = 0; k < DDIM; k += 4) {
        v2f a = *(const v2f*)(Arow + k + kb);             // global_load_b64
        const float* Br0 = B + (size_t)(k + kb) * DDIM;   // row kb
        const float* Br1 = Br0 + DDIM;                    // row kb+1
        v2f b0 = { Br0[n0      + nc], Br1[n0      + nc] };
        v2f b1 = { Br0[n0 + 16 + nc], Br1[n0 + 16 + nc] };
        v2f b2 = { Br0[n0 + 32 + nc], Br1[n0 + 32 + nc] };
        v2f b3 = { Br0[n0 + 48 + nc], Br1[n0 + 48 + nc] };
        acc0 = __builtin_amdgcn_wmma_f32_16x16x4_f32(false, a, false, b0, (short)0, acc0, false, false);
        acc1 = __builtin_amdgcn_wmma_f32_16x16x4_f32(false, a, false, b1, (short)0, acc1, false, false);
        acc2 = __builtin_amdgcn_wmma_f32_16x16x4_f32(false, a, false, b2, (short)0, acc2, false, false);
        acc3 = __builtin_amdgcn_wmma_f32_16x16x4_f32(false, a, false, b3, (short)0, acc3, false, false);
    }

    const int mb = m0 + (lane >> 4) * 8;
#define STORE_TILE(ACC, T)                                                     \
    {                                                                          \
        const int n = n0 + (T)*16 + nc;                                        \
        const float bv = bias ? bias[n] : 0.0f;                                \
        _Pragma("unroll")                                                      \
        for (int v = 0; v < 8; ++v)                                            \
            C[(size_t)(mb + v) * DDIM + n] = ACC[v] + bv;                      \
    }
    STORE_TILE(acc0, 0)
    STORE_TILE(acc1, 1)
    STORE_TILE(acc2, 2)
    STORE_TILE(acc3, 3)
#undef STORE_TILE
}

// ---------------------------------------------------------------------------
// Kernel 2: fused per-edge logit + weighted scatter-add.
// One wave32 per edge:
//   a_e = dot(q[s], k[r])                       (coalesced, L2-resident rows)
//   out[r][:] += a_e * h_proj[s][:]             (hw f32 atomics, L2-resident)
// ---------------------------------------------------------------------------
__global__ __launch_bounds__(256)
void edge_scatter_kernel(const float* __restrict__ hp,
                         const float* __restrict__ q,
                         const float* __restrict__ kmat,
                         const int* __restrict__ senders,
                         const int* __restrict__ receivers,
                         float* __restrict__ out, int E) {
    const int e = blockIdx.x * (blockDim.x >> 5) + (threadIdx.x >> 5);
    const int lane = threadIdx.x & 31;
    if (e >= E) return;

    const int s = senders[e];
    const int r = receivers[e];

    const float* qs = q    + (size_t)s * DDIM;
    const float* kr = kmat + (size_t)r * DDIM;

    float acc = 0.0f;
#pragma unroll
    for (int i = 0; i < DDIM / 32; ++i)
        acc += qs[lane + i * 32] * kr[lane + i * 32];

    // wave32 butterfly reduction -> all lanes hold a_e
#pragma unroll
    for (int off = 16; off > 0; off >>= 1)
        acc += __shfl_xor(acc, off, 32);

    const float* hs = hp + (size_t)s * DDIM;
    float* orow = out + (size_t)r * DDIM;
#pragma unroll
    for (int i = 0; i < DDIM / 32; ++i)
        unsafeAtomicAdd(&orow[lane + i * 32], acc * hs[lane + i * 32]);
}

// ---------------------------------------------------------------------------
// Kernel 3: in-place ReLU
// ---------------------------------------------------------------------------
__global__ __launch_bounds__(256)
void relu_kernel(float* __restrict__ out, int n) {
    int i = blockIdx.x * blockDim.x + threadIdx.x;
    if (i < n) out[i] = fmaxf(out[i], 0.0f);
}

// ---------------------------------------------------------------------------
extern "C" void kernel_launch(void* const* d_in, const int* in_sizes, int n_in,
                              void* d_out, int out_size, void* d_ws, size_t ws_size,
                              hipStream_t stream) {
    const float* h  = (const float*)d_in[0];
    const float* W  = (const float*)d_in[1];
    const float* Wq = (const float*)d_in[2];
    const float* bq = (const float*)d_in[3];
    const float* Wk = (const float*)d_in[4];
    const float* bk = (const float*)d_in[5];
    const int* senders   = (const int*)d_in[6];
    const int* receivers = (const int*)d_in[7];
    float* out = (float*)d_out;
    float* ws  = (float*)d_ws;   // [h_proj | q | k], 3 * 8 MB

    const int E = in_sizes[6];
    const int nOut = NNODES * DDIM;

    // 0) zero output (scatter target)
    zero_kernel<<<(nOut + 255) / 256, 256, 0, stream>>>(out, nOut);

    // 1) h_proj, q, k via f32 WMMA (grid: 4 n-strips x 512 m-tiles x 3 matrices)
    {
        dim3 grid(DDIM / 64, NNODES / 16, 3);
        gemm3_wmma_f32<<<grid, 32, 0, stream>>>(h, W, Wq, bq, Wk, bk, ws);
    }

    // 2) per-edge logit + scatter (8 edges per 256-thread block)
    {
        const float* hp = ws;
        const float* q  = ws + (size_t)NNODES * DDIM;
        const float* k  = ws + 2 * (size_t)NNODES * DDIM;
        int blocks = (E + 7) / 8;
        edge_scatter_kernel<<<blocks, 256, 0, stream>>>(hp, q, k, senders, receivers, out, E);
    }

    // 3) ReLU
    relu_kernel<<<(nOut + 255) / 256, 256, 0, stream>>>(out, nOut);
}